// flow_reg_4329327034898
// MI455X (gfx1250) — compile-verified
//
#include <hip/hip_runtime.h>
#include <math.h>
#include <stdint.h>

// Problem constants (from reference): x is (4, 9, 9, 256, 448) f32.
#define MD     4
#define UDIM   9
#define VDIM   9
#define NC     (UDIM * VDIM)   // 81 displacement channels
#define HGT    256
#define WID    448
#define PLANE  (HGT * WID)     // 114688 pixels per (b) plane
#define BATCH  4
#define WSZ    3               // truncation window radius

#define TILE   128             // pixels per workgroup (divides PLANE)
#define RS     (TILE + 16)     // LDS row stride in dwords (16-DW pad => bank-safe)
#define NCP    84              // channels padded to multiple of 4 for K=4 WMMA
#define NCHUNK (NCP / 4)       // 21 WMMA K-chunks

typedef float v2f  __attribute__((ext_vector_type(2)));
typedef float v8f  __attribute__((ext_vector_type(8)));
typedef unsigned int u32x4 __attribute__((ext_vector_type(4)));
typedef int   i32x4 __attribute__((ext_vector_type(4)));
typedef int   i32x8 __attribute__((ext_vector_type(8)));

// Weight of reduction column n for channel c: [1, fx, fy, 0, ..., 0].
__device__ __forceinline__ float wf(int c, int n) {
  const float fx = (float)(c / VDIM - MD);
  const float fy = (float)(c % VDIM - MD);
  float r = (n == 2) ? fy : 0.0f;
  r = (n == 1) ? fx : r;
  r = (n == 0) ? 1.0f : r;
  return r;
}

__global__ __launch_bounds__(TILE)
void flowreg_kernel(const float* __restrict__ x, float* __restrict__ out) {
  __shared__ float xs[NCP * RS];       // staged tile: 84 rows x 144 dwords
  __shared__ float sums[TILE][4];      // per-pixel [Zl, Fx, Fy] from WMMA
  __shared__ v2f   Btab[NCHUNK * 32];  // per-lane B operands, all 21 chunks

  const int t      = threadIdx.x;
  const int base   = (int)(blockIdx.x) * TILE;          // flat pixel base
  const int b      = base / PLANE;                      // tiles never cross b
  const int ppbase = base % PLANE;
  const float* gx  = x + (size_t)b * NC * PLANE + ppbase;

  // ---- Stage tile (81 channels x 128 pixels) into LDS via the TDM ----
#if defined(__has_builtin) && __has_builtin(__builtin_amdgcn_tensor_load_to_lds)
  if (t < 32) {  // wave 0 issues the tensor DMA for the whole workgroup
    const uint32_t lds = (uint32_t)(uintptr_t)&xs[0];
    const uint64_t ga  = (uint64_t)(uintptr_t)gx;
    u32x4 g0;
    g0[0] = 1u;                                            // count=1, user D#
    g0[1] = lds;                                           // lds_addr
    g0[2] = (uint32_t)(ga & 0xFFFFFFFFu);                  // global_addr lo
    g0[3] = (uint32_t)((ga >> 32) & 0x01FFFFFFu) | (2u << 30);  // hi + type=2
    i32x8 g1;
    g1[0] = (int)((2u << 16)      // data_size = 4 bytes
                | (1u << 20)      // pad_enable
                | (6u << 22)      // pad_interval: every 128 DWORDs
                | (15u << 25));   // pad_amount: 16 DWORDs -> row stride 144
    g1[1] = (int)((uint32_t)TILE << 16);   // tensor_dim0 = 128 (tile-exact)
    g1[2] = (int)((uint32_t)NC   << 16);   // tensor_dim1 = 81
    g1[3] = (int)((uint32_t)TILE << 16);   // tile_dim0   = 128
    g1[4] = (int)NC;                       // tile_dim1 = 81, tile_dim2 = 0
    g1[5] = (int)PLANE;                    // tensor_dim0_stride = h*w
    g1[6] = 0;
    g1[7] = 0;
    i32x4 gz = {0, 0, 0, 0};
#if __clang_major__ >= 23
    i32x8 gz8 = {0, 0, 0, 0, 0, 0, 0, 0};
    __builtin_amdgcn_tensor_load_to_lds(g0, g1, gz, gz, gz8, 0);
#else
    __builtin_amdgcn_tensor_load_to_lds(g0, g1, gz, gz, 0);
#endif
  }
#endif

  // ---- Fill the B-operand table while the TDM transfer is in flight ----
  // Btab[k][lane] = { wf(4k + 2*(lane>>4) + 0, lane&15),
  //                   wf(4k + 2*(lane>>4) + 1, lane&15) }  (one-time cost)
  for (int idx = t; idx < NCHUNK * 32; idx += TILE) {
    const int k  = idx >> 5;
    const int ln = idx & 31;
    const int c0 = 4 * k + 2 * (ln >> 4);
    const int n  = ln & 15;
    v2f w;
    w.x = wf(c0 + 0, n);
    w.y = wf(c0 + 1, n);
    Btab[idx] = w;
  }

#if defined(__has_builtin) && __has_builtin(__builtin_amdgcn_tensor_load_to_lds)
  if (t < 32) __builtin_amdgcn_s_wait_tensorcnt(0);
#else
  for (int c = 0; c < NC; ++c)
    xs[c * RS + t] = gx[(size_t)c * PLANE + t];
#endif
  __syncthreads();

  // ---- Pass 1 (LDS): max + argmax over the 81 channels of pixel t ----
  float m = -3.402823466e38f;
  int   bc = 0;
#pragma unroll
  for (int c = 0; c < NC; ++c) {
    const float v = xs[c * RS + t];
    if (v > m) { m = v; bc = c; }
  }
  const int bu = bc / VDIM, bv = bc % VDIM;

  // ---- Pass 2 (LDS): exp, global sums, masked sums; store masked e back ----
  // Note: argmax lies inside its own 7x7 mask => local max == global max m.
  float Zg = 0.0f, Sg = 0.0f, Sl = 0.0f;
  int u = 0, v = 0;
#pragma unroll
  for (int c = 0; c < NC; ++c) {
    const float xv = xs[c * RS + t];
    const float e  = __expf(xv - m);
    Zg += e;
    Sg += e * xv;
    int du = u - bu; du = du < 0 ? -du : du;
    int dv = v - bv; dv = dv < 0 ? -dv : dv;
    const bool in = (du <= WSZ) && (dv <= WSZ);
    Sl += in ? e * xv : 0.0f;
    xs[c * RS + t] = in ? e : 0.0f;   // masked e, consumed by WMMA below
    if (++v == VDIM) { v = 0; ++u; }
  }
  xs[81 * RS + t] = 0.0f;             // zero K-padding rows
  xs[82 * RS + t] = 0.0f;
  xs[83 * RS + t] = 0.0f;
  __syncthreads();

  // ---- WMMA reduction: (16 pixels x 84) @ (84 x [1, fx, fy, 0...]) ----
  // A (16x4 f32): lanes 0-15 hold M=0..15 with K={0,1}; lanes 16-31 K={2,3}.
  // B (4x16 f32): VGPR j holds row K = j + 2*half, column N = lane&15.
  // Both 16-pixel groups of this wave share each B load (two accumulators).
  const int lane   = t & 31;
  const int waveid = t >> 5;
  const int half   = lane >> 4;
  const int mrow   = lane & 15;
  const int P0     = waveid * 32;       // first 16-pixel group base
  const int P1     = P0 + 16;           // second 16-pixel group base

  v8f acc0 = {0.f, 0.f, 0.f, 0.f, 0.f, 0.f, 0.f, 0.f};
  v8f acc1 = {0.f, 0.f, 0.f, 0.f, 0.f, 0.f, 0.f, 0.f};
#pragma unroll
  for (int k = 0; k < NCHUNK; ++k) {
    const int c0 = 4 * k + 2 * half;
    const v2f Bv = Btab[k * 32 + lane];           // ds_load_b64, imm offset
    v2f A0, A1;
    A0.x = xs[(c0 + 0) * RS + P0 + mrow];
    A0.y = xs[(c0 + 1) * RS + P0 + mrow];
    A1.x = xs[(c0 + 0) * RS + P1 + mrow];
    A1.y = xs[(c0 + 1) * RS + P1 + mrow];
    acc0 = __builtin_amdgcn_wmma_f32_16x16x4_f32(
        false, A0, false, Bv, (short)0, acc0, false, false);
    acc1 = __builtin_amdgcn_wmma_f32_16x16x4_f32(
        false, A1, false, Bv, (short)0, acc1, false, false);
  }
  // D layout: lane n (0-15) = column N=n; VGPR r = row M=r (+8 for hi half).
  if (mrow < 3) {
#pragma unroll
    for (int r = 0; r < 8; ++r) {
      sums[P0 + r + 8 * half][mrow] = acc0[r];
      sums[P1 + r + 8 * half][mrow] = acc1[r];
    }
  }
  __syncthreads();

  // ---- Finalize pixel t ----
  const float Zl = sums[t][0];
  const float Fx = sums[t][1];
  const float Fy = sums[t][2];
  const float invZl = 1.0f / Zl;
  const float flow_x = Fx * invZl;
  const float flow_y = Fy * invZl;
  // entropy = m + log Z - (sum e*x)/Z ; normalizers 1/log(49), 1/log(81)
  const float lent = (m + __logf(Zl) - Sl * invZl) * 0.25694940f;
  const float gent = (m + __logf(Zg) - Sg / Zg)    * 0.22756112f;

  const int pp = ppbase + t;
  float* flow = out;                                   // (b,2,h,w)
  float* ent  = out + (size_t)BATCH * 2 * PLANE;       // (b,2,h,w)
  flow[(size_t)b * 2 * PLANE + pp]         = flow_x;
  flow[(size_t)b * 2 * PLANE + PLANE + pp] = flow_y;
  ent [(size_t)b * 2 * PLANE + pp]         = lent;
  ent [(size_t)b * 2 * PLANE + PLANE + pp] = gent;
}

extern "C" void kernel_launch(void* const* d_in, const int* in_sizes, int n_in,
                              void* d_out, int out_size, void* d_ws, size_t ws_size,
                              hipStream_t stream) {
  (void)in_sizes; (void)n_in; (void)out_size; (void)d_ws; (void)ws_size;
  const float* x = (const float*)d_in[0];
  float* out = (float*)d_out;
  const int nblocks = (BATCH * PLANE) / TILE;  // 3584 workgroups of 128 (4 waves)
  flowreg_kernel<<<nblocks, TILE, 0, stream>>>(x, out);
}